// LabelPairwiseLoss_40767829573855
// MI455X (gfx1250) — compile-verified
//
#include <hip/hip_runtime.h>
#include <math.h>

typedef __attribute__((ext_vector_type(2))) float v2f;
typedef __attribute__((ext_vector_type(8))) float v8f;

#define THR_LO   0.6f
#define THR_HI   0.8f
#define D_FEAT   64
#define CLAMP_V  100.0f

// Per-edge loss terms, accumulated into per-lane partial sums.
__device__ __forceinline__ void edge_terms(float d, float pi, float pj,
                                           float& s_sim, float& s_dis,
                                           float& n_sim, float& n_dis) {
    const bool m_sim = (pi >= THR_HI) && (pj >= THR_HI);
    const bool m_dis = ((pi >= THR_HI) && (pj < THR_LO)) ||
                       ((pj >= THR_HI) && (pi < THR_LO));
    const float p = expf(-d);
    // bce_sim = -max(log(p), -100); mirrors reference's exp->log round trip
    const float bce_sim = fminf(-logf(p), CLAMP_V);
    // bce_disim = -max(log1p(-p), -100)
    const float bce_dis = fminf(-log1pf(-p), CLAMP_V);
    if (m_sim) { s_sim += bce_sim; n_sim += 1.0f; }
    if (m_dis) { s_dis += bce_dis; n_dis += 1.0f; }
}

// One wave handles 16 edges per group.
// Lane L (half h = L>>4, e = L&15) holds Diff[e][4c + 2h + {0,1}] for chunk c.
// This is simultaneously the A (16x4) and B (4x16) register layout of
// V_WMMA_F32_16X16X4_F32, so wmma(diff, diff, acc) accumulates G = Diff*Diff^T;
// the diagonal of the 16x16 f32 accumulator is d^2 per edge (exact f32 math).
__global__ void pair_loss_wmma(const int*   __restrict__ edges,
                               const float* __restrict__ probas,
                               const float* __restrict__ feats,
                               float*       __restrict__ partials,
                               int nGroups) {
    const int lane   = threadIdx.x & 31;
    const int wave   = (int)((blockIdx.x * blockDim.x + threadIdx.x) >> 5);
    const int nWaves = (int)((gridDim.x * blockDim.x) >> 5);
    const int e_in_g = lane & 15;
    const int half   = lane >> 4;          // 0 or 1

    float s_sim = 0.0f, s_dis = 0.0f, n_sim = 0.0f, n_dis = 0.0f;

    for (int g = wave; g < nGroups; g += nWaves) {   // wave-uniform: EXEC all-1s at WMMA
        const int base = g << 4;
        const int i = edges[2 * (base + e_in_g) + 0];
        const int j = edges[2 * (base + e_in_g) + 1];
        // Prefetch the next group's edge segment (global_prefetch_b8).
        __builtin_prefetch(&edges[2 * (base + 16) + 2 * e_in_g], 0, 0);

        const float* fa = feats + (size_t)i * D_FEAT + 2 * half;
        const float* fb = feats + (size_t)j * D_FEAT + 2 * half;

        v2f diffs[16];
#pragma unroll
        for (int c = 0; c < 16; ++c) {
            v2f a = *(const v2f*)(fa + 4 * c);   // 8B loads; lanes e / e+16 cover
            v2f b = *(const v2f*)(fb + 4 * c);   // disjoint halves of each row
            diffs[c] = a - b;
        }

        v8f acc = {};
#pragma unroll
        for (int c = 0; c < 16; ++c) {
            // D = A*B + C with A = B = diff chunk  =>  acc += Diff*Diff^T
            acc = __builtin_amdgcn_wmma_f32_16x16x4_f32(
                      false, diffs[c], false, diffs[c],
                      (short)0, acc, false, false);
        }

        // Diagonal extraction: element (e,e) lives in acc[e%8] on lane e (e<8)
        // or lane e+16 (e>=8). Owner lanes are 0..7 (edges 0..7) and 24..31
        // (edges 8..15); each owner's own (i,j) already matches its edge.
        float dsq = 0.0f;
#pragma unroll
        for (int v = 0; v < 8; ++v) {
            if (lane == v || lane == v + 24) dsq = acc[v];
        }
        const bool owner = (lane < 8) || (lane >= 24);
        if (owner) {
            const float pi = probas[i];
            const float pj = probas[j];
            edge_terms(sqrtf(dsq), pi, pj, s_sim, s_dis, n_sim, n_dis);
        }
    }

    // Wave reduction (wave32), then one deterministic partial per wave.
#pragma unroll
    for (int off = 16; off > 0; off >>= 1) {
        s_sim += __shfl_xor(s_sim, off, 32);
        s_dis += __shfl_xor(s_dis, off, 32);
        n_sim += __shfl_xor(n_sim, off, 32);
        n_dis += __shfl_xor(n_dis, off, 32);
    }
    if (lane == 0) {
        partials[4 * wave + 0] = s_sim;
        partials[4 * wave + 1] = s_dis;
        partials[4 * wave + 2] = n_sim;
        partials[4 * wave + 3] = n_dis;
    }
}

// Single-wave finalize: fixed-order reduction of per-wave partials (deterministic,
// no float atomics), scalar tail edges, then the weighted loss scalar.
__global__ void pair_loss_final(const float* __restrict__ partials, int nWaves,
                                const int*   __restrict__ edges,
                                const float* __restrict__ probas,
                                const float* __restrict__ feats,
                                int tailStart, int nEdges,
                                float* __restrict__ out) {
    const int lane = threadIdx.x & 31;
    float s_sim = 0.0f, s_dis = 0.0f, n_sim = 0.0f, n_dis = 0.0f;

    for (int w = lane; w < nWaves; w += 32) {
        s_sim += partials[4 * w + 0];
        s_dis += partials[4 * w + 1];
        n_sim += partials[4 * w + 2];
        n_dis += partials[4 * w + 3];
    }
#pragma unroll
    for (int off = 16; off > 0; off >>= 1) {
        s_sim += __shfl_xor(s_sim, off, 32);
        s_dis += __shfl_xor(s_dis, off, 32);
        n_sim += __shfl_xor(n_sim, off, 32);
        n_dis += __shfl_xor(n_dis, off, 32);
    }

    if (lane == 0) {
        for (int e = tailStart; e < nEdges; ++e) {   // < 16 edges; usually zero
            const int i = edges[2 * e + 0];
            const int j = edges[2 * e + 1];
            float dsq = 0.0f;
            for (int k = 0; k < D_FEAT; ++k) {
                const float df = feats[(size_t)i * D_FEAT + k]
                               - feats[(size_t)j * D_FEAT + k];
                dsq += df * df;
            }
            edge_terms(sqrtf(dsq), probas[i], probas[j],
                       s_sim, s_dis, n_sim, n_dis);
        }
        const float total = n_sim + n_dis;
        const float pos_w = n_dis / total;
        const float neg_w = n_sim / total;
        out[0] = (s_sim * pos_w + s_dis * neg_w) / total;
    }
}

extern "C" void kernel_launch(void* const* d_in, const int* in_sizes, int n_in,
                              void* d_out, int out_size, void* d_ws, size_t ws_size,
                              hipStream_t stream) {
    const int*   edges  = (const int*)d_in[0];    // (N_EDGES, 2) indices
    const float* probas = (const float*)d_in[1];  // (N_NODES,)
    const float* feats  = (const float*)d_in[2];  // (N_NODES, 64)
    float* out = (float*)d_out;
    float* partials = (float*)d_ws;

    const int nEdges  = in_sizes[0] / 2;
    const int nGroups = nEdges / 16;
    const int tail    = nGroups * 16;

    const int THREADS = 256;                       // 8 waves/block
    // Enough waves for latency hiding, clamped by work and by workspace size.
    int blocks = 1024;
    const int wantBlocks = (nGroups + 7) / 8;      // >= 1 group per wave
    if (wantBlocks < blocks) blocks = wantBlocks;
    const int maxWavesWs = (int)(ws_size / (4 * sizeof(float)));
    if (maxWavesWs < blocks * 8) blocks = maxWavesWs / 8;
    if (blocks < 1) blocks = 1;
    const int nWaves = blocks * (THREADS / 32);

    pair_loss_wmma<<<blocks, THREADS, 0, stream>>>(edges, probas, feats,
                                                   partials, nGroups);
    pair_loss_final<<<1, 32, 0, stream>>>(partials, nWaves, edges, probas, feats,
                                          tail, nEdges, out);
}